// DimensionAdaptivePoolingForSensors_2138893713803
// MI455X (gfx1250) — compile-verified
//
#include <hip/hip_runtime.h>
#include <hip/hip_bf16.h>

// Problem constants (fixed by the reference):
//   x: (B=64, W_IN=512, H_IN=48, M=64) fp32, row-major
//   out: (B, PW=16, PH=6, M) fp32 flattened -> 64 x 6144
// Bins are exact: 512/16 = 32 (WBIN), 48/6 = 8 (HBIN).
#define B_SZ   64
#define W_IN   512
#define H_IN   48
#define M_CH   64
#define PW     16
#define PH     6
#define WBIN   32
#define HBIN   8

typedef float v4f __attribute__((ext_vector_type(4)));

// One thread per float4 of the output.
// g = ((b*PW + iwb)*PH + ihb)*16 + m4  -> out4[g] = result.
// Total threads = 64*16*6*16 = 98304 = 384 blocks x 256 threads.
__global__ __launch_bounds__(256) void dap_pool_kernel(const float* __restrict__ x,
                                                       float* __restrict__ out) {
    const int g = blockIdx.x * blockDim.x + threadIdx.x;

    const int m4  = g & 15;          // float4 index within M (m = 4*m4)
    const int ihb = (g >> 4) % PH;   // height bin
    const int t   = g / (PH * 16);   // b*PW + iwb
    const int iwb = t & (PW - 1);
    const int b   = t >> 4;

    // Base element index in float4 units:
    //   e4 = ((b*W_IN + iwb*WBIN)*H_IN + ihb*HBIN) * (M_CH/4) + m4
    const int base4 = ((b * W_IN + iwb * WBIN) * H_IN + ihb * HBIN) * (M_CH / 4) + m4;
    const v4f* __restrict__ x4 = reinterpret_cast<const v4f*>(x) + base4;

    // Strides in float4 units
    const int iw_stride = H_IN * (M_CH / 4);  // 48*16 = 768
    const int ih_stride = (M_CH / 4);         // 16

    v4f acc;
    acc[0] = acc[1] = acc[2] = acc[3] = -__builtin_inff();

    #pragma unroll 2
    for (int iw = 0; iw < WBIN; ++iw) {
        const v4f* __restrict__ p = x4 + iw * iw_stride;
        // Keep the HBM stream ahead: prefetch this thread's next iw-slice
        // (lowers to global_prefetch_b8 on gfx1250).
        __builtin_prefetch((const void*)(p + iw_stride), 0, 1);
        #pragma unroll
        for (int ih = 0; ih < HBIN; ++ih) {
            // Streamed exactly once -> non-temporal: don't thrash WGP$/L2.
            // Lowers to global_load_b128 ... th:TH_LOAD_NT
            const v4f v = __builtin_nontemporal_load(p + ih * ih_stride);
            acc = __builtin_elementwise_max(acc, v);
        }
    }

    // Output is written once and not re-read by this kernel -> NT store.
    __builtin_nontemporal_store(acc, reinterpret_cast<v4f*>(out) + g);
}

extern "C" void kernel_launch(void* const* d_in, const int* in_sizes, int n_in,
                              void* d_out, int out_size, void* d_ws, size_t ws_size,
                              hipStream_t stream) {
    (void)in_sizes; (void)n_in; (void)d_ws; (void)ws_size; (void)out_size;
    const float* x = reinterpret_cast<const float*>(d_in[0]);
    float* out = reinterpret_cast<float*>(d_out);

    const int total4 = B_SZ * PW * PH * (M_CH / 4);  // 98304 float4 outputs
    const int block = 256;
    const int grid = total4 / block;                  // 384
    dap_pool_kernel<<<grid, block, 0, stream>>>(x, out);
}